// ComplexEMA_71846212927931
// MI455X (gfx1250) — compile-verified
//
#include <hip/hip_runtime.h>
#include <hip/hip_bf16.h>
#include <math.h>

typedef __attribute__((ext_vector_type(8))) float v8f;
typedef __attribute__((ext_vector_type(2))) float v2f;
typedef __attribute__((ext_vector_type(4))) float f4;

#define Dd 1024
#define Nn 16
#define Bb 8
#define Ll 4096
#define CH 256     // Ll / 16 chunks
#define TILES 16   // CH / 16

// compiler-only reordering fence: single-wave workgroup, LDS ops are
// hardware-ordered within a wave, so no s_barrier / counter drain is needed.
#define CBAR() asm volatile("" ::: "memory")

// workspace layout (floats)
#define OFF_PRE  0                         // P real  [D][16 modes][16 t]
#define OFF_PIM  (Dd*256)                  // P imag
#define OFF_GRE  (2*Dd*256)                // G real  [D][16 j][16 n]
#define OFF_GIM  (3*Dd*256)                // G imag
#define OFF_KT   (4*Dd*256)                // k taps  [D][16]
#define OFF_Q16R (4*Dd*256 + Dd*16)        // q^16 re [D][16]
#define OFF_Q16I (4*Dd*256 + 2*Dd*16)      // q^16 im [D][16]

// ---------------------------------------------------------------------------
// Kernel 1: per-channel constants.
//   P[n][t]   = q_n^(15-t)                 (for A = P @ X chunk carries)
//   G[j][n]   = g_n * q_n^(j+1)            (cross-chunk projection)
//   k[m]      = Re( sum_n g_n q_n^m )      (16-tap triangular Toeplitz)
//   q16[n]    = q_n^16                     (chunk-to-chunk decay)
// ---------------------------------------------------------------------------
__global__ void __launch_bounds__(256) cema_setup(
    const float* __restrict__ p_logit, const float* __restrict__ lqr_,
    const float* __restrict__ lqi_, const float* __restrict__ gre_,
    const float* __restrict__ gim_, float* __restrict__ ws) {
  const int d = blockIdx.x;
  const int tid = threadIdx.x;
  const int n = tid & 15;        // mode
  const int j = tid >> 4;        // position / tap / exponent slot
  const int dn = d * 16 + n;

  const float p = 1.0f / (1.0f + __expf(-p_logit[dn]));
  const float lqr = lqr_[dn];
  const float lqi = lqi_[dn];
  const float s = 0.25f * p;                 // SCALE = sqrt(1/16)
  const float gr = gre_[dn] * s;
  const float gi = gim_[dn] * s;

  // P[n][t] = q^(15-t), with t = j
  {
    const float e = (float)(15 - j);
    const float m = __expf(e * lqr);
    ws[OFF_PRE + d * 256 + n * 16 + j] = m * __cosf(e * lqi);
    ws[OFF_PIM + d * 256 + n * 16 + j] = m * __sinf(e * lqi);
  }
  // w = q^(j+1) ;  G[j][n] = g * w ;  at j==15, w == q^16
  {
    const float e = (float)(j + 1);
    const float m = __expf(e * lqr);
    const float wr = m * __cosf(e * lqi);
    const float wi = m * __sinf(e * lqi);
    ws[OFF_GRE + d * 256 + j * 16 + n] = gr * wr - gi * wi;
    ws[OFF_GIM + d * 256 + j * 16 + n] = gr * wi + gi * wr;
    if (j == 15) {
      ws[OFF_Q16R + dn] = wr;
      ws[OFF_Q16I + dn] = wi;
    }
  }
  // k[m] = Re( sum_n g_n q_n^m ), m = j ; reduce over n through LDS
  __shared__ float red[16][17];
  {
    const float e = (float)j;
    const float m = __expf(e * lqr);
    red[j][n] = gr * (m * __cosf(e * lqi)) - gi * (m * __sinf(e * lqi));
  }
  __syncthreads();
  if (n == 0) {
    float acc = 0.0f;
#pragma unroll
    for (int q = 0; q < 16; ++q) acc += red[j][q];
    ws[OFF_KT + d * 16 + j] = acc;
  }
}

// ---------------------------------------------------------------------------
// Kernel 2: one wave32 per (b,d) row.  Per 16-chunk tile (fused pipeline):
//   A_tile = P @ X_tile                      (8  x v_wmma_f32_16x16x4_f32)
//   scan   : C[c] = q16*C[c-1] + A[c-1]      (16 lanes, complex, tiny LDS tile)
//   Y_tile = T0@X + Gr@Cr - Gi@Ci            (12 x v_wmma_f32_16x16x4_f32)
//            + omega * x residual, b128 stores
// x row staged global->LDS with GLOBAL_LOAD_ASYNC_TO_LDS_B128 (ASYNCcnt),
// waited in two phases (async loads complete in order) so the first 8 tiles
// overlap the tail of the HBM copy.  No workgroup barriers: single wave,
// LDS is wave-ordered -> only compiler fences, global stores stay in flight.
// ---------------------------------------------------------------------------
__global__ void __launch_bounds__(32) cema_conv(
    const float* __restrict__ x, const float* __restrict__ omega,
    const float* __restrict__ ws, float* __restrict__ out) {
  __shared__ __attribute__((aligned(16))) float xs[Ll];    // 16 KB
  __shared__ float Ctr[16 * 17];   // C tile real [mode][chunk_local] (+pad)
  __shared__ float Cti[16 * 17];   // C tile imag

  const int bd = blockIdx.x;
  const int d = bd & (Dd - 1);
  const int ln = threadIdx.x;
  const int half = ln >> 4;      // 0: lanes 0-15, 1: lanes 16-31
  const int lm = ln & 15;

  const float* __restrict__ xrow = x + (size_t)bd * Ll;
  float* __restrict__ orow = out + (size_t)bd * Ll;

  // ---- async stage x row into LDS: 32 x b128 per lane, offset hits both
  //      the LDS and global address (ISA: dsaddr/memaddr both + INST_OFFSET)
  {
    const unsigned lds0 = (unsigned)(uintptr_t)(void*)xs + (unsigned)(ln * 16);
    const unsigned long long ga = (unsigned long long)(uintptr_t)xrow +
                                  (unsigned long long)(ln * 16);
#pragma unroll
    for (int i = 0; i < 32; ++i) {
      asm volatile("global_load_async_to_lds_b128 %0, %1, off offset:%2"
                   :
                   : "v"(lds0), "v"(ga), "i"(i * 512)
                   : "memory");
    }
  }

  // ---- tile-invariant A-matrix fragments (16x4 f32 layout per ISA),
  //      loaded while the async LDS copy is in flight ----
  const float* __restrict__ Pre = ws + OFF_PRE + d * 256;
  const float* __restrict__ Pim = ws + OFF_PIM + d * 256;
  const float* __restrict__ Gre = ws + OFF_GRE + d * 256;
  const float* __restrict__ Gim = ws + OFF_GIM + d * 256;
  const float* __restrict__ ktap = ws + OFF_KT + d * 16;

  v2f aPr[4], aPi[4], aT[4], aGr[4], aGiN[4];
#pragma unroll
  for (int s = 0; s < 4; ++s) {
    const int k0 = 4 * s + 2 * half;   // K indices (k0, k0+1) for this lane
    aPr[s].x = Pre[lm * 16 + k0];
    aPr[s].y = Pre[lm * 16 + k0 + 1];
    aPi[s].x = Pim[lm * 16 + k0];
    aPi[s].y = Pim[lm * 16 + k0 + 1];
    aGr[s].x = Gre[lm * 16 + k0];
    aGr[s].y = Gre[lm * 16 + k0 + 1];
    aGiN[s].x = -Gim[lm * 16 + k0];            // fold the minus of -Gi@Ci
    aGiN[s].y = -Gim[lm * 16 + k0 + 1];
    aT[s].x = (lm >= k0) ? ktap[lm - k0] : 0.0f;        // lower-tri Toeplitz
    aT[s].y = (lm >= k0 + 1) ? ktap[lm - k0 - 1] : 0.0f;
  }

  // per-mode chunk decay + running complex carry (lanes 0-15 active in scan)
  float q16r = 0.0f, q16i = 0.0f;
  if (ln < 16) {
    q16r = ws[OFF_Q16R + d * 16 + ln];
    q16i = ws[OFF_Q16I + d * 16 + ln];
  }
  float cr = 0.0f, ci = 0.0f;
  const float om = omega[d];

  auto tile_body = [&](int i) {
    // ---- B fragments of X tile (4x16 slices, K = intra-chunk t) ----
    v2f bX[4];
#pragma unroll
    for (int s = 0; s < 4; ++s) {
      const int t0 = 4 * s + 2 * half;
      bX[s].x = xs[(16 * i + lm) * 16 + t0];
      bX[s].y = xs[(16 * i + lm) * 16 + t0 + 1];
    }

    // ---- A_tile = P @ X_tile (complex -> two f32 accumulators) ----
    v8f accr = {};
    v8f acci = {};
#pragma unroll
    for (int s = 0; s < 4; ++s) {
      accr = __builtin_amdgcn_wmma_f32_16x16x4_f32(false, aPr[s], false, bX[s],
                                                   (short)0, accr, false, false);
      acci = __builtin_amdgcn_wmma_f32_16x16x4_f32(false, aPi[s], false, bX[s],
                                                   (short)0, acci, false, false);
    }
    // D layout: VGPR v -> M=v(+8 hi lanes)=mode, N=lane&15=chunk_local
#pragma unroll
    for (int v = 0; v < 8; ++v) {
      Ctr[(v + 8 * half) * 17 + lm] = accr[v];
      Cti[(v + 8 * half) * 17 + lm] = acci[v];
    }
    CBAR();

    // ---- complex carry scan across this tile's 16 chunks (in place) ----
    if (ln < 16) {
      for (int c = 0; c < 16; ++c) {
        const float ar = Ctr[ln * 17 + c];
        const float ai = Cti[ln * 17 + c];
        Ctr[ln * 17 + c] = cr;
        Cti[ln * 17 + c] = ci;
        const float nr = q16r * cr - q16i * ci + ar;
        const float ni = q16r * ci + q16i * cr + ai;
        cr = nr;
        ci = ni;
      }
    }
    CBAR();

    // ---- Y_tile = T0@X + Gr@Cr - Gi@Ci ----
    v2f bCr[4], bCi[4];
#pragma unroll
    for (int s = 0; s < 4; ++s) {
      const int t0 = 4 * s + 2 * half;     // K = mode index here
      bCr[s].x = Ctr[t0 * 17 + lm];
      bCr[s].y = Ctr[(t0 + 1) * 17 + lm];
      bCi[s].x = Cti[t0 * 17 + lm];
      bCi[s].y = Cti[(t0 + 1) * 17 + lm];
    }
    v8f acc = {};
#pragma unroll
    for (int s = 0; s < 4; ++s)
      acc = __builtin_amdgcn_wmma_f32_16x16x4_f32(false, aT[s], false, bX[s],
                                                  (short)0, acc, false, false);
#pragma unroll
    for (int s = 0; s < 4; ++s)
      acc = __builtin_amdgcn_wmma_f32_16x16x4_f32(false, aGr[s], false, bCr[s],
                                                  (short)0, acc, false, false);
#pragma unroll
    for (int s = 0; s < 4; ++s)
      acc = __builtin_amdgcn_wmma_f32_16x16x4_f32(false, aGiN[s], false, bCi[s],
                                                  (short)0, acc, false, false);

    // l = (16*i + lm)*16 + 8*half + v  (v = 0..7 contiguous -> two b128 stores)
    const int base = (16 * i + lm) * 16 + 8 * half;
    f4 lo, hi;
    lo.x = acc[0] + om * xs[base + 0];
    lo.y = acc[1] + om * xs[base + 1];
    lo.z = acc[2] + om * xs[base + 2];
    lo.w = acc[3] + om * xs[base + 3];
    hi.x = acc[4] + om * xs[base + 4];
    hi.y = acc[5] + om * xs[base + 5];
    hi.z = acc[6] + om * xs[base + 6];
    hi.w = acc[7] + om * xs[base + 7];
    *(f4*)(orow + base) = lo;
    *(f4*)(orow + base + 4) = hi;
    CBAR();   // C tile reused next iteration (wave-ordered LDS, no s_barrier)
  };

  // async loads complete in order: first 16 b128s cover tiles 0-7.
  asm volatile("s_wait_asynccnt 0x10" ::: "memory");
  for (int i = 0; i < TILES / 2; ++i) tile_body(i);
  asm volatile("s_wait_asynccnt 0x0" ::: "memory");
  for (int i = TILES / 2; i < TILES; ++i) tile_body(i);
}

extern "C" void kernel_launch(void* const* d_in, const int* in_sizes, int n_in,
                              void* d_out, int out_size, void* d_ws, size_t ws_size,
                              hipStream_t stream) {
  const float* x       = (const float*)d_in[0];
  const float* p_logit = (const float*)d_in[1];
  const float* lqr     = (const float*)d_in[2];
  const float* lqi     = (const float*)d_in[3];
  const float* gre     = (const float*)d_in[4];
  const float* gim     = (const float*)d_in[5];
  const float* omega   = (const float*)d_in[6];
  float* out = (float*)d_out;
  float* ws  = (float*)d_ws;   // needs 4*D*256 + 3*D*16 floats ~= 4.4 MB

  cema_setup<<<Dd, 256, 0, stream>>>(p_logit, lqr, lqi, gre, gim, ws);
  cema_conv<<<Bb * Dd, 32, 0, stream>>>(x, omega, ws, out);
}